// tokenizer_29618094474254
// MI455X (gfx1250) — compile-verified
//
#include <hip/hip_runtime.h>
#include <stdint.h>

// ---------------------------------------------------------------------------
// out[b,g,:] = gene_table[g,:] + (mask[g] ? mut_table[8,:] : mut_table[X[b,g],:])
// B=32, G=20000, F=256, fp32. Pure streaming-store problem (~655 MB written,
// ~23 MB read) -> bound by HBM write stream: ~29us at 23.3 TB/s. No matrix op,
// so WMMA is deliberately not used. CDNA5 paths: async global->LDS staging of
// mut_table, s_wait_asynccnt, b128 vector IO, non-temporal output stores.
// ---------------------------------------------------------------------------

typedef float v4f __attribute__((ext_vector_type(4)));
typedef int   v4i __attribute__((vector_size(16)));   // matches builtin proto

#define G_NUM   20000
#define F_NUM   256
#define VOCAB   9
#define BATCH   32
#define GPB     8      // genes per block (20000 / 8 = 2500 blocks exactly)
#define THREADS 256    // 8 waves (wave32)

#if defined(__gfx1250__) && \
    __has_builtin(__builtin_amdgcn_global_load_async_to_lds_b128) && \
    __has_builtin(__builtin_amdgcn_s_wait_asynccnt)
#define ASYNC_LDS_OK 1
typedef __attribute__((address_space(1))) v4i* gptr_v4i;
typedef __attribute__((address_space(3))) v4i* lptr_v4i;
#endif

// ---------------- mask setup (tiny, runs once per launch) ------------------
__global__ void zero_mask_kernel(uint32_t* __restrict__ mask_words, int nwords) {
  int i = blockIdx.x * blockDim.x + threadIdx.x;
  if (i < nwords) mask_words[i] = 0u;
}

__global__ void scatter_mask_kernel(const int* __restrict__ geneset, int n,
                                    uint8_t* __restrict__ mask) {
  int i = blockIdx.x * blockDim.x + threadIdx.x;
  if (i < n) {
    int g = geneset[i];
    if (g >= 0 && g < G_NUM) mask[g] = (uint8_t)1;
  }
}

// ---------------- main fused kernel ----------------------------------------
// Thread mapping: c = tid & 63 -> float4 chunk along F (64 lanes * 16B = 1KB
// contiguous row), bq = tid >> 6 -> handles batches bq*8 .. bq*8+7.
__global__ __launch_bounds__(THREADS)
void tokenizer_fused_kernel(const int*     __restrict__ X,          // [B,G]
                            const float*   __restrict__ gene_table, // [G,F]
                            const float*   __restrict__ mut_table,  // [VOCAB,F]
                            const uint8_t* __restrict__ mask,       // [G]
                            float*         __restrict__ out)        // [B,G,F]
{
  __shared__ float lds_mut[VOCAB * F_NUM];   // 9216 B: whole mut_table
  __shared__ int   lds_tok[GPB * BATCH];     // 1024 B: token ids for this block

  const int tid = threadIdx.x;
  const int g0  = blockIdx.x * GPB;

  // Stage mut_table (9x256 fp32 = 576 x b128) into LDS via async DMA path.
#if defined(ASYNC_LDS_OK)
  for (int i = tid; i < (VOCAB * F_NUM) / 4; i += THREADS) {
    __builtin_amdgcn_global_load_async_to_lds_b128(
        (gptr_v4i)(uintptr_t)(mut_table + i * 4),   // AS1 int4*
        (lptr_v4i)(&lds_mut[i * 4]),                // AS3 int4*
        /*offset=*/0, /*cpol=*/0);
  }
#else
  for (int i = tid; i < (VOCAB * F_NUM) / 4; i += THREADS) {
    ((v4f*)lds_mut)[i] = ((const v4f*)mut_table)[i];
  }
#endif

  // Stage the 8*32 token ids: tid -> b = tid>>3, gi = tid&7 (exactly 256).
  {
    const int b = tid >> 3, gi = tid & 7;
    lds_tok[gi * BATCH + b] = X[(size_t)b * G_NUM + g0 + gi];
  }

#if defined(ASYNC_LDS_OK)
  __builtin_amdgcn_s_wait_asynccnt(0);   // s_wait_asynccnt 0
#endif
  __syncthreads();

  const int c  = tid & 63;   // float4 chunk index within feature dim
  const int bq = tid >> 6;   // which group of 8 batches

  // All 8 gene-row chunks up front: one base, 1KB immediate-offset strides.
  // gene_table is L2-resident (20.5 MB vs 192 MB L2) and reused 32x.
  v4f gene[GPB];
  {
    const float* gbase = gene_table + (size_t)g0 * F_NUM + c * 4;
#pragma unroll
    for (int gi = 0; gi < GPB; ++gi)
      gene[gi] = *(const v4f*)(gbase + gi * F_NUM);
  }

  // 8 mask bytes in one aligned 64-bit load (g0 % 8 == 0, d_ws is aligned).
  const uint64_t m8 = *(const uint64_t*)(mask + g0);

#pragma unroll
  for (int bb = 0; bb < 8; ++bb) {
    const int b = bq * 8 + bb;
    // One 64-bit address per batch; the 8 stores differ by 1KB -> ioffset.
    float* obase = out + ((size_t)b * G_NUM + g0) * F_NUM + c * 4;
#pragma unroll
    for (int gi = 0; gi < GPB; ++gi) {
      const bool is_masked = ((m8 >> (8 * gi)) & 0xffu) != 0;
      const int  tok = is_masked ? (VOCAB - 1) : lds_tok[gi * BATCH + b];
      const v4f  mv  = *(const v4f*)(&lds_mut[tok * F_NUM + c * 4]);
      const v4f  r   = gene[gi] + mv;
      // Write-once 655 MB stream: non-temporal so L2 keeps gene_table hot.
      __builtin_nontemporal_store(r, (v4f*)(obase + (size_t)gi * F_NUM));
    }
  }
}

// ---------------- launch ---------------------------------------------------
extern "C" void kernel_launch(void* const* d_in, const int* in_sizes, int n_in,
                              void* d_out, int out_size, void* d_ws, size_t ws_size,
                              hipStream_t stream) {
  // setup_inputs order:
  //  0: X_converted     int32 [B,G]
  //  1: mask_percentage int32 scalar (unused by reference math)
  //  2: test_geneset    int32 [N_MASK]
  //  3: gene_table      fp32  [G,F]
  //  4: mut_table       fp32  [VOCAB,F]
  const int*   X          = (const int*)  d_in[0];
  const int*   geneset    = (const int*)  d_in[2];
  const float* gene_table = (const float*)d_in[3];
  const float* mut_table  = (const float*)d_in[4];
  float*       out        = (float*)      d_out;

  uint8_t* mask   = (uint8_t*)d_ws;          // G_NUM bytes of scratch
  const int nmask = in_sizes[2];

  const int mask_words = (G_NUM + 3) / 4;
  zero_mask_kernel<<<(mask_words + 255) / 256, 256, 0, stream>>>(
      (uint32_t*)mask, mask_words);
  scatter_mask_kernel<<<(nmask + 255) / 256, 256, 0, stream>>>(
      geneset, nmask, mask);
  tokenizer_fused_kernel<<<G_NUM / GPB, THREADS, 0, stream>>>(
      X, gene_table, mut_table, mask, out);
}